// ContrastiveLoss_69561290326699
// MI455X (gfx1250) — compile-verified
//
#include <hip/hip_runtime.h>
#include <hip/hip_bf16.h>
#include <stddef.h>

// Problem constants from the reference: B=4096, D=512, t=0.2 (1/t = 5).
#define NROWS 4096
#define TWO_N 8192
#define DDIM 512
#define INV_T 5.0f
#define LP 516   // LDS row pitch (floats): 516*4B rows are 16B-aligned, bank-shift 4 -> conflict-free

typedef float v2f __attribute__((ext_vector_type(2)));
typedef float v8f __attribute__((ext_vector_type(8)));

// ---------------------------------------------------------------------------
// Kernel 1: L2-normalize rows of [emb_i; emb_j] into Z, and store
// exp(dot(z_i, z_k)/t) per row (for denom_fu).
// ---------------------------------------------------------------------------
__global__ __launch_bounds__(256) void k_normalize_fu(
    const float* __restrict__ emb_i, const float* __restrict__ emb_j,
    const float* __restrict__ emb_k, float* __restrict__ Z,
    float* __restrict__ fuexp) {
  __shared__ float r0[8], r1[8], r2[8];
  const int b = blockIdx.x;
  const int t = threadIdx.x;
  const bool isI = (b < NROWS);
  const float* src = isI ? (emb_i + (size_t)b * DDIM)
                         : (emb_j + (size_t)(b - NROWS) * DDIM);
  const float2 x = ((const float2*)src)[t];          // 256 thr * 2 = 512 elems
  float ss = x.x * x.x + x.y * x.y;
  float ssk = 0.0f, dik = 0.0f;
  if (isI) {
    const float2 y = ((const float2*)(emb_k + (size_t)b * DDIM))[t];
    ssk = y.x * y.x + y.y * y.y;
    dik = x.x * y.x + x.y * y.y;
  }
#pragma unroll
  for (int o = 16; o > 0; o >>= 1) {
    ss  += __shfl_xor(ss, o, 32);
    ssk += __shfl_xor(ssk, o, 32);
    dik += __shfl_xor(dik, o, 32);
  }
  if ((t & 31) == 0) { r0[t >> 5] = ss; r1[t >> 5] = ssk; r2[t >> 5] = dik; }
  __syncthreads();
  float tot = 0.0f;
#pragma unroll
  for (int w = 0; w < 8; ++w) tot += r0[w];
  const float ni = fmaxf(sqrtf(tot), 1e-12f);
  const float inv = 1.0f / ni;
  float2 z; z.x = x.x * inv; z.y = x.y * inv;
  ((float2*)(Z + (size_t)b * DDIM))[t] = z;
  if (isI && t == 0) {
    float totk = 0.0f, totd = 0.0f;
    for (int w = 0; w < 8; ++w) { totk += r1[w]; totd += r2[w]; }
    const float nk = fmaxf(sqrtf(totk), 1e-12f);
    fuexp[b] = __expf((totd / (ni * nk)) * INV_T);
  }
}

// ---------------------------------------------------------------------------
// Kernel 2: diag[r] = dot(Z_r, Z_r) ; pos[r] = dot(Z_r, Z_{r+N}) for r < N.
// ---------------------------------------------------------------------------
__global__ __launch_bounds__(256) void k_diag_pos(
    const float* __restrict__ Z, float* __restrict__ diag,
    float* __restrict__ pos) {
  __shared__ float r0[8], r1[8];
  const int b = blockIdx.x, t = threadIdx.x;
  const float2 x = ((const float2*)(Z + (size_t)b * DDIM))[t];
  float dd = x.x * x.x + x.y * x.y;
  float pp = 0.0f;
  if (b < NROWS) {
    const float2 y = ((const float2*)(Z + (size_t)(b + NROWS) * DDIM))[t];
    pp = x.x * y.x + x.y * y.y;
  }
#pragma unroll
  for (int o = 16; o > 0; o >>= 1) {
    dd += __shfl_xor(dd, o, 32);
    pp += __shfl_xor(pp, o, 32);
  }
  if ((t & 31) == 0) { r0[t >> 5] = dd; r1[t >> 5] = pp; }
  __syncthreads();
  if (t == 0) {
    float td = 0.0f, tp = 0.0f;
    for (int w = 0; w < 8; ++w) { td += r0[w]; tp += r1[w]; }
    diag[b] = td;
    if (b < NROWS) pos[b] = tp;
  }
}

// ---------------------------------------------------------------------------
// Kernel 3: fused GEMM + row-wise exp-sum on v_wmma_f32_16x16x4_f32.
// Grid: 256 blocks = 128 row-blocks (64 rows) x 2 column halves (4096 cols).
// Block = 256 threads (8 wave32). Wave (ry 0..3, cx 0..1) computes a 16x32
// output slab (two C tiles): one A fragment feeds two WMMAs per K-step
// (1.5 LDS b64 loads per WMMA instead of 2 -> LDS-bandwidth relief).
// rowSpart[half][r] = sum_{c in half} exp(sim[r][c]/t)  (diag included).
// ---------------------------------------------------------------------------
__global__ __launch_bounds__(256) void k_gemm_rowsum(
    const float* __restrict__ Z, float* __restrict__ rowSpart) {
  extern __shared__ float smem[];
  float* ldsA = smem;                 // 64 * LP floats (132 KB)
  float* ldsB = smem + 64 * LP;       // 64 * LP floats (132 KB)
  float* ldsS = smem + 128 * LP;      // 128 floats: [cx][row_local]
  const int t = threadIdx.x;
  const int lane = t & 31;
  const int wave = t >> 5;
  const int ry = wave >> 1;           // 0..3: row tile (16 rows each)
  const int cx = wave & 1;            // 0..1: 32-column slab
  const int hi = lane >> 4;           // lanes 16-31 hold K+2..K+3
  const int rb = blockIdx.x >> 1;     // row block (64 rows)
  const int ch = blockIdx.x & 1;      // column half (4096 cols)
  const int row0 = rb * 64;
  const int col0 = ch * NROWS;

  // Stage A: 64 rows of Z (float4 global loads -> 16B-aligned LDS rows).
  for (int i = t; i < 64 * 128; i += 256) {
    const int r = i >> 7, c4 = i & 127;
    const float4 v = ((const float4*)(Z + (size_t)(row0 + r) * DDIM))[c4];
    *(float4*)(ldsA + r * LP + c4 * 4) = v;
  }

  const float* Arow  = ldsA + (ry * 16 + (lane & 15)) * LP + 2 * hi;

  float racc0[8], racc1[8];
#pragma unroll
  for (int j = 0; j < 8; ++j) { racc0[j] = 0.0f; racc1[j] = 0.0f; }

  for (int chunk = 0; chunk < 64; ++chunk) {
    const int n0 = col0 + chunk * 64;
    __syncthreads();                              // ldsB consumed by prev iter
    for (int i = t; i < 64 * 128; i += 256) {
      const int r = i >> 7, c4 = i & 127;
      const float4 v = ((const float4*)(Z + (size_t)(n0 + r) * DDIM))[c4];
      *(float4*)(ldsB + r * LP + c4 * 4) = v;
    }
    __syncthreads();
    if (chunk + 1 < 64 && t < 64)                 // hint next chunk into caches
      __builtin_prefetch(Z + (size_t)(n0 + 64 + t) * DDIM, 0, 0);

    const float* Brow0 = ldsB + (cx * 32 + (lane & 15)) * LP + 2 * hi;
    const float* Brow1 = Brow0 + 16 * LP;
    v8f acc0 = {}, acc1 = {};
#pragma unroll 8
    for (int k = 0; k < DDIM; k += 4) {
      const v2f a  = *(const v2f*)(Arow  + k);    // A fragment: reused twice
      const v2f b0 = *(const v2f*)(Brow0 + k);
      const v2f b1 = *(const v2f*)(Brow1 + k);
      acc0 = __builtin_amdgcn_wmma_f32_16x16x4_f32(
          false, a, false, b0, (short)0, acc0, false, false);
      acc1 = __builtin_amdgcn_wmma_f32_16x16x4_f32(
          false, a, false, b1, (short)0, acc1, false, false);
    }
#pragma unroll
    for (int j = 0; j < 8; ++j) {
      racc0[j] += __expf(acc0[j] * INV_T);
      racc1[j] += __expf(acc1[j] * INV_T);
    }
  }

  // Row sums: C layout -> VGPR j holds row (ry*16 + j + 8*hi), col = lane&15.
#pragma unroll
  for (int j = 0; j < 8; ++j) {
    float v = racc0[j] + racc1[j];                // both tiles, same row
    v += __shfl_xor(v, 1, 16);
    v += __shfl_xor(v, 2, 16);
    v += __shfl_xor(v, 4, 16);
    v += __shfl_xor(v, 8, 16);
    if ((lane & 15) == 0) ldsS[cx * 64 + ry * 16 + 8 * hi + j] = v;
  }
  __syncthreads();
  if (t < 64) {                                   // fixed-order, deterministic
    const float S = ldsS[t] + ldsS[64 + t];
    rowSpart[(size_t)ch * TWO_N + row0 + t] = S;
  }
}

// ---------------------------------------------------------------------------
// Kernel 4: single-block deterministic finalize.
// fu = 2*sum(fuexp);
// loss = mean_r( log(rowSpart0[r]+rowSpart1[r] - exp(diag*5) + fu) - 5*pos ).
// ---------------------------------------------------------------------------
__global__ __launch_bounds__(256) void k_finalize(
    const float* __restrict__ rowSpart, const float* __restrict__ diag,
    const float* __restrict__ pos, const float* __restrict__ fuexp,
    float* __restrict__ out) {
  __shared__ float red[8];
  const int t = threadIdx.x;
  float f = 0.0f;
  for (int i = t; i < NROWS; i += 256) f += fuexp[i];
#pragma unroll
  for (int o = 16; o > 0; o >>= 1) f += __shfl_xor(f, o, 32);
  if ((t & 31) == 0) red[t >> 5] = f;
  __syncthreads();
  float fu = 0.0f;
  for (int w = 0; w < 8; ++w) fu += red[w];
  fu *= 2.0f;
  __syncthreads();

  float acc = 0.0f;
  for (int r = t; r < TWO_N; r += 256) {
    const float S = rowSpart[r] + rowSpart[TWO_N + r];
    const float denom = S - __expf(diag[r] * INV_T) + fu;
    acc += __logf(denom) - pos[r & (NROWS - 1)] * INV_T;
  }
#pragma unroll
  for (int o = 16; o > 0; o >>= 1) acc += __shfl_xor(acc, o, 32);
  if ((t & 31) == 0) red[t >> 5] = acc;
  __syncthreads();
  if (t == 0) {
    float s = 0.0f;
    for (int w = 0; w < 8; ++w) s += red[w];
    out[0] = s / (float)TWO_N;
  }
}

// ---------------------------------------------------------------------------
extern "C" void kernel_launch(void* const* d_in, const int* in_sizes, int n_in,
                              void* d_out, int out_size, void* d_ws,
                              size_t ws_size, hipStream_t stream) {
  (void)in_sizes; (void)n_in; (void)out_size; (void)ws_size;
  const float* emb_i = (const float*)d_in[0];
  const float* emb_j = (const float*)d_in[1];
  const float* emb_k = (const float*)d_in[2];
  float* out = (float*)d_out;

  // Workspace layout (floats): Z[8192*512] | diag[8192] | pos[4096]
  //                            | fuexp[4096] | rowSpart[2*8192]  (~16.2 MB)
  float* Z        = (float*)d_ws;
  float* diag     = Z + (size_t)TWO_N * DDIM;
  float* pos      = diag + TWO_N;
  float* fuexp    = pos + NROWS;
  float* rowSpart = fuexp + NROWS;

  k_normalize_fu<<<TWO_N, 256, 0, stream>>>(emb_i, emb_j, emb_k, Z, fuexp);
  k_diag_pos<<<TWO_N, 256, 0, stream>>>(Z, diag, pos);

  const size_t smem = (size_t)(128 * LP + 128) * sizeof(float); // ~264.7 KB < 320 KB/WGP
  k_gemm_rowsum<<<256, 256, smem, stream>>>(Z, rowSpart);

  k_finalize<<<1, 256, 0, stream>>>(rowSpart, diag, pos, fuexp, out);
}